// GAT_15358803051066
// MI455X (gfx1250) — compile-verified
//
#include <hip/hip_runtime.h>
#include <math.h>

// ---------------------------------------------------------------------------
// GAT layer for MI455X (gfx1250, wave32).
//  - fp32 WMMA (v_wmma_f32_16x16x4_f32) for all GEMMs (precision-true path)
//  - A-panels held in registers (64 VGPRs) and reused across weight matrices
//  - global_prefetch_b8 to warm L2 for weight panels
//  - edge softmax via 3 recompute passes (node arrays are L2-resident; never
//    materialize E x 128 tensors that would spill the 192MB L2)
//  - fused residual+LN+FFN(512)+PReLU+residual+LN epilogue per 16-row tile
// ---------------------------------------------------------------------------

typedef __attribute__((ext_vector_type(2))) float v2f;
typedef __attribute__((ext_vector_type(8))) float v8f;

#define FDIM   128
#define FH     512
#define LN_EPS 1e-5f
#define SQRT_DH 5.656854249492380f   /* sqrt(128/4) */

__device__ __forceinline__ v8f wmma4(v2f a, v2f b, v8f c) {
  // D = A(16x4,f32) * B(4x16,f32) + C(16x16,f32)
  return __builtin_amdgcn_wmma_f32_16x16x4_f32(
      /*neg_a=*/false, a, /*neg_b=*/false, b,
      /*c_mod=*/(short)0, c, /*reuse_a=*/false, /*reuse_b=*/false);
}

// A fragment: row-major A, 16 rows, K block [k0,k0+4)
// lanes 0-15: row=lane, K={k0,k0+1}; lanes 16-31: row=lane-16, K={k0+2,k0+3}
__device__ __forceinline__ v2f loadA(const float* A, int lda, int k0, int lane) {
  int half = lane >> 4, row = lane & 15;
  const float* p = A + row * lda + k0 + half * 2;
  v2f a; a.x = p[0]; a.y = p[1];
  return a;
}

// B fragment: row-major B (K x Ncols, ldb), cols [base,base+16), K block k0
// lanes 0-15: K={k0,k0+1}; lanes 16-31: K={k0+2,k0+3}; col=lane&15
__device__ __forceinline__ v2f loadB(const float* B, int ldb, int k0, int lane) {
  int half = lane >> 4, col = lane & 15;
  const float* p = B + (k0 + half * 2) * ldb + col;
  v2f b; b.x = p[0]; b.y = p[ldb];
  return b;
}

// ---------------------------------------------------------------------------
// Q/K/V projections: one block per 16-row tile; A panel (16x128) cached in
// 64 VGPRs per wave and swept across Wq/Wk/Wv -> feat read from HBM once.
// ---------------------------------------------------------------------------
__global__ void __launch_bounds__(256)
qkv_kernel(const float* __restrict__ feat,
           const float* __restrict__ Wq, const float* __restrict__ Wk,
           const float* __restrict__ Wv,
           float* __restrict__ q, float* __restrict__ k, float* __restrict__ v,
           int N) {
  __shared__ float at[16 * FDIM];                       // 8 KB
  const int rb = blockIdx.x * 16;

  for (int i = threadIdx.x; i < 16 * FDIM; i += 256)
    at[i] = feat[(size_t)rb * FDIM + i];

  const int wave = threadIdx.x >> 5, lane = threadIdx.x & 31;
  const int cb = wave * 16;                             // this wave's 16 cols
  const int half = lane >> 4, col = lane & 15;

  // warm L2 for the weight panels (global_prefetch_b8)
  __builtin_prefetch(Wq + (size_t)lane * FDIM + cb, 0, 1);
  __builtin_prefetch(Wk + (size_t)lane * FDIM + cb, 0, 1);
  __builtin_prefetch(Wv + (size_t)lane * FDIM + cb, 0, 1);

  __syncthreads();

  // cache the full A panel: 32 k-blocks x v2f = 64 VGPRs
  v2f af[32];
#pragma unroll
  for (int kk = 0; kk < 32; ++kk)
    af[kk] = loadA(at, FDIM, kk * 4, lane);

  const float* Ws[3]  = {Wq, Wk, Wv};
  float*       outs[3] = {q, k, v};
#pragma unroll
  for (int z = 0; z < 3; ++z) {
    const float* W = Ws[z];
    v8f c = {};
#pragma unroll
    for (int kk = 0; kk < 32; ++kk)
      c = wmma4(af[kk], loadB(W + cb, FDIM, kk * 4, lane), c);
    float* out = outs[z];
#pragma unroll
    for (int r = 0; r < 8; ++r)
      out[(size_t)(rb + r + half * 8) * FDIM + cb + col] = c[r];
  }
}

// ---------------------------------------------------------------------------
// segment-max encoding: order-preserving float->u32 map for atomicMax
// ---------------------------------------------------------------------------
__device__ __forceinline__ unsigned encf(float f) {
  unsigned u = __float_as_uint(f);
  return (u & 0x80000000u) ? ~u : (u | 0x80000000u);
}
__device__ __forceinline__ float decf(unsigned u) {
  return __uint_as_float((u & 0x80000000u) ? (u ^ 0x80000000u) : ~u);
}

__global__ void init_kernel(unsigned* __restrict__ mbits,
                            float* __restrict__ den, float* __restrict__ agg,
                            int NF) {
  int i = blockIdx.x * 256 + threadIdx.x;
  if (i < NF) {
    mbits[i] = 0x007FFFFFu;   // encf(-inf)
    den[i] = 0.f;
    agg[i] = 0.f;
  }
}

// ---------------------------------------------------------------------------
// Edge passes: one wave32 per edge, float4 per lane (16B/lane, coalesced).
// e = (k[src] + v[dst]) * sqrt(Dh) is recomputed each pass (L2-resident gathers)
// ---------------------------------------------------------------------------
__global__ void __launch_bounds__(256)
edge_max_kernel(const float* __restrict__ kmat, const float* __restrict__ vmat,
                const int* __restrict__ src, const int* __restrict__ dst,
                unsigned* __restrict__ mbits, int E) {
  int e = blockIdx.x * 8 + (threadIdx.x >> 5);
  if (e >= E) return;
  int lane = threadIdx.x & 31;
  int s = src[e], d = dst[e];
  float4 k4 = reinterpret_cast<const float4*>(kmat + (size_t)s * FDIM)[lane];
  float4 v4 = reinterpret_cast<const float4*>(vmat + (size_t)d * FDIM)[lane];
  unsigned* mp = mbits + (size_t)d * FDIM + lane * 4;
  atomicMax(mp + 0, encf((k4.x + v4.x) * SQRT_DH));
  atomicMax(mp + 1, encf((k4.y + v4.y) * SQRT_DH));
  atomicMax(mp + 2, encf((k4.z + v4.z) * SQRT_DH));
  atomicMax(mp + 3, encf((k4.w + v4.w) * SQRT_DH));
}

__global__ void __launch_bounds__(256)
edge_sum_kernel(const float* __restrict__ kmat, const float* __restrict__ vmat,
                const int* __restrict__ src, const int* __restrict__ dst,
                const unsigned* __restrict__ mbits, float* __restrict__ den,
                int E) {
  int e = blockIdx.x * 8 + (threadIdx.x >> 5);
  if (e >= E) return;
  int lane = threadIdx.x & 31;
  int s = src[e], d = dst[e];
  float4 k4 = reinterpret_cast<const float4*>(kmat + (size_t)s * FDIM)[lane];
  float4 v4 = reinterpret_cast<const float4*>(vmat + (size_t)d * FDIM)[lane];
  const unsigned* mp = mbits + (size_t)d * FDIM + lane * 4;
  float* dp = den + (size_t)d * FDIM + lane * 4;
  atomicAdd(dp + 0, __expf((k4.x + v4.x) * SQRT_DH - decf(mp[0])));
  atomicAdd(dp + 1, __expf((k4.y + v4.y) * SQRT_DH - decf(mp[1])));
  atomicAdd(dp + 2, __expf((k4.z + v4.z) * SQRT_DH - decf(mp[2])));
  atomicAdd(dp + 3, __expf((k4.w + v4.w) * SQRT_DH - decf(mp[3])));
}

__global__ void __launch_bounds__(256)
edge_agg_kernel(const float* __restrict__ kmat, const float* __restrict__ vmat,
                const float* __restrict__ qmat,
                const int* __restrict__ src, const int* __restrict__ dst,
                const unsigned* __restrict__ mbits, const float* __restrict__ den,
                float* __restrict__ agg, int E) {
  int e = blockIdx.x * 8 + (threadIdx.x >> 5);
  if (e >= E) return;
  int lane = threadIdx.x & 31;
  int s = src[e], d = dst[e];
  float4 k4 = reinterpret_cast<const float4*>(kmat + (size_t)s * FDIM)[lane];
  float4 v4 = reinterpret_cast<const float4*>(vmat + (size_t)d * FDIM)[lane];
  float4 q4 = reinterpret_cast<const float4*>(qmat + (size_t)s * FDIM)[lane];
  const unsigned* mp = mbits + (size_t)d * FDIM + lane * 4;
  const float* dp = den + (size_t)d * FDIM + lane * 4;
  float* ap = agg + (size_t)d * FDIM + lane * 4;
  atomicAdd(ap + 0, q4.x * __expf((k4.x + v4.x) * SQRT_DH - decf(mp[0])) / dp[0]);
  atomicAdd(ap + 1, q4.y * __expf((k4.y + v4.y) * SQRT_DH - decf(mp[1])) / dp[1]);
  atomicAdd(ap + 2, q4.z * __expf((k4.z + v4.z) * SQRT_DH - decf(mp[2])) / dp[2]);
  atomicAdd(ap + 3, q4.w * __expf((k4.w + v4.w) * SQRT_DH - decf(mp[3])) / dp[3]);
}

// ---------------------------------------------------------------------------
// Fused epilogue per 16-row tile:
//   x  = LN(agg + feat)
//   h  = PReLU(x @ W1 + b1)        (WMMA, A panel register-cached)
//   y  = x + (h @ W2 + b2)         (WMMA)
//   out= LN(y)
// ---------------------------------------------------------------------------
__global__ void __launch_bounds__(256)
post_kernel(const float* __restrict__ feat, const float* __restrict__ agg,
            const float* __restrict__ g, const float* __restrict__ bln,
            const float* __restrict__ W1, const float* __restrict__ b1,
            const float* __restrict__ alpha,
            const float* __restrict__ W2, const float* __restrict__ b2,
            float* __restrict__ out, int N) {
  __shared__ float xln[16 * FDIM];  // 8 KB
  __shared__ float h[16 * FH];      // 32 KB
  __shared__ float y[16 * FDIM];    // 8 KB
  const int rb = blockIdx.x * 16;
  const int tid = threadIdx.x;
  const int wave = tid >> 5, lane = tid & 31;
  const int half = lane >> 4, col = lane & 15;

  // warm L2 for this wave's weight panels (global_prefetch_b8)
  __builtin_prefetch(W1 + (size_t)lane * FH + wave * 64, 0, 1);
  __builtin_prefetch(W2 + (size_t)lane * FDIM + wave * 16, 0, 1);

  // ---- Stage A: LN(agg + feat); 16 threads per row, shfl_xor reduce ----
  {
    int row = tid >> 4, c0 = tid & 15;
    float s = 0.f, s2 = 0.f, vals[8];
#pragma unroll
    for (int j = 0; j < 8; ++j) {
      int cc = c0 + j * 16;
      float x = agg[(size_t)(rb + row) * FDIM + cc] +
                feat[(size_t)(rb + row) * FDIM + cc];
      vals[j] = x; s += x; s2 += x * x;
    }
#pragma unroll
    for (int m = 1; m < 16; m <<= 1) {       // stays inside 16-lane group
      s  += __shfl_xor(s,  m);
      s2 += __shfl_xor(s2, m);
    }
    float mean = s * (1.f / FDIM);
    float inv  = rsqrtf(s2 * (1.f / FDIM) - mean * mean + LN_EPS);
#pragma unroll
    for (int j = 0; j < 8; ++j) {
      int cc = c0 + j * 16;
      xln[row * FDIM + cc] = (vals[j] - mean) * inv * g[cc] + bln[cc];
    }
  }
  __syncthreads();

  // ---- Stage B: h = PReLU(xln @ W1 + b1); A panel cached in registers,
  //      reused across this wave's 4 hidden column tiles [64w, 64w+64) ----
  {
    v2f af[32];
#pragma unroll
    for (int kk = 0; kk < 32; ++kk)
      af[kk] = loadA(xln, FDIM, kk * 4, lane);

    for (int t = 0; t < 4; ++t) {
      int cb = wave * 64 + t * 16;
      v8f c = {};
#pragma unroll
      for (int kk = 0; kk < 32; ++kk)
        c = wmma4(af[kk], loadB(W1 + cb, FH, kk * 4, lane), c);
      float bb = b1[cb + col], al = alpha[cb + col];
#pragma unroll
      for (int r = 0; r < 8; ++r) {
        float hv = c[r] + bb;
        h[(r + half * 8) * FH + cb + col] = (hv > 0.f) ? hv : al * hv;
      }
    }
  }
  __syncthreads();

  // ---- Stage C: y = xln + h @ W2 + b2, wave w -> out cols [16w,16w+16) ----
  {
    int cb = wave * 16;
    v8f c = {};
    for (int k0 = 0; k0 < FH; k0 += 4) {
      v2f a = loadA(h, FH, k0, lane);
      v2f b = loadB(W2 + cb, FDIM, k0, lane);
      c = wmma4(a, b, c);
    }
    float bb = b2[cb + col];
#pragma unroll
    for (int r = 0; r < 8; ++r) {
      int row = r + half * 8;
      y[row * FDIM + cb + col] = xln[row * FDIM + cb + col] + c[r] + bb;
    }
  }
  __syncthreads();

  // ---- Stage D: out = LN(y) ----
  {
    int row = tid >> 4, c0 = tid & 15;
    float s = 0.f, s2 = 0.f, vals[8];
#pragma unroll
    for (int j = 0; j < 8; ++j) {
      float x = y[row * FDIM + c0 + j * 16];
      vals[j] = x; s += x; s2 += x * x;
    }
#pragma unroll
    for (int m = 1; m < 16; m <<= 1) {
      s  += __shfl_xor(s,  m);
      s2 += __shfl_xor(s2, m);
    }
    float mean = s * (1.f / FDIM);
    float inv  = rsqrtf(s2 * (1.f / FDIM) - mean * mean + LN_EPS);
#pragma unroll
    for (int j = 0; j < 8; ++j) {
      int cc = c0 + j * 16;
      out[(size_t)(rb + row) * FDIM + cc] = (vals[j] - mean) * inv * g[cc] + bln[cc];
    }
  }
}

// ---------------------------------------------------------------------------
extern "C" void kernel_launch(void* const* d_in, const int* in_sizes, int n_in,
                              void* d_out, int out_size, void* d_ws, size_t ws_size,
                              hipStream_t stream) {
  const float* feat = (const float*)d_in[0];
  const float* Wq   = (const float*)d_in[1];
  const float* Wk   = (const float*)d_in[2];
  const float* Wv   = (const float*)d_in[3];
  const float* ln_g = (const float*)d_in[4];
  const float* ln_b = (const float*)d_in[5];
  const float* W1   = (const float*)d_in[6];
  const float* b1   = (const float*)d_in[7];
  const float* alph = (const float*)d_in[8];
  const float* W2   = (const float*)d_in[9];
  const float* b2   = (const float*)d_in[10];
  const int*   src  = (const int*)d_in[11];
  const int*   dst  = (const int*)d_in[12];

  const int N  = in_sizes[0] / FDIM;   // 50000
  const int E  = in_sizes[11];         // 800000
  const size_t NF = (size_t)N * FDIM;

  // workspace layout: q | k | v | mbits(u32) | den | agg  (6 * N * F * 4B)
  float*    q     = (float*)d_ws;
  float*    k     = q + NF;
  float*    v     = k + NF;
  unsigned* mbits = (unsigned*)(v + NF);
  float*    den   = (float*)(mbits + NF);
  float*    agg   = den + NF;

  qkv_kernel<<<(N + 15) / 16, 256, 0, stream>>>(feat, Wq, Wk, Wv, q, k, v, N);

  init_kernel<<<(int)((NF + 255) / 256), 256, 0, stream>>>(mbits, den, agg, (int)NF);

  int eb = (E + 7) / 8;   // 8 edges (waves) per 256-thread block
  edge_max_kernel<<<eb, 256, 0, stream>>>(k, v, src, dst, mbits, E);
  edge_sum_kernel<<<eb, 256, 0, stream>>>(k, v, src, dst, mbits, den, E);
  edge_agg_kernel<<<eb, 256, 0, stream>>>(k, v, q, src, dst, mbits, den, agg, E);

  post_kernel<<<(N + 15) / 16, 256, 0, stream>>>(feat, agg, ln_g, ln_b,
                                                 W1, b1, alph, W2, b2,
                                                 (float*)d_out, N);
}